// MyModel_87522843559153
// MI455X (gfx1250) — compile-verified
//
#include <hip/hip_runtime.h>
#include <hip/hip_bf16.h>

// CDNA5 / gfx1250, wave32.
// Streaming conv1d (K=5, CIN=32, COUT=64, stride=2) unrolled into a GEMM:
//   y[4+2m, :] = inputs[64*m .. 64*m+159] (contig fp32) x W[160,64]
// computed with V_WMMA_F32_16X16X4_F32 (fp32 in/out, matches reference).
// The GEMM epilogue also writes the interleaved zero rows (odd t), so the
// output is produced in a single sweep; a tiny helper kernel handles rows
// 0..3 and the flags array.

typedef float v2f __attribute__((ext_vector_type(2)));
typedef float v8f __attribute__((ext_vector_type(8)));

#define CIN   32
#define COUT  64
#define KTOT  160           // 5 * 32
#define ROWS_PER_BLOCK 128  // 4 outer iters * 2 m-tiles * 16

// ---------------------------------------------------------------------------
// Zero head rows (t < 4; or everything if zero_all) and write flags for all t.
// flag[t] = (t >= 4) && (t % 2 == 0)
// ---------------------------------------------------------------------------
__global__ __launch_bounds__(256) void fill_head_and_flags(float* __restrict__ out,
                                                           float* __restrict__ flags,
                                                           int T, int zero_all) {
    const size_t stride = (size_t)gridDim.x * blockDim.x;
    const size_t gid    = (size_t)blockIdx.x * blockDim.x + threadIdx.x;

    const int head_rows = zero_all ? T : (T < 4 ? T : 4);
    float4* out4 = (float4*)out;
    const size_t nvec = (size_t)head_rows * (COUT / 4);
    for (size_t i = gid; i < nvec; i += stride) {
        out4[i] = make_float4(0.f, 0.f, 0.f, 0.f);
    }
    for (size_t t = gid; t < (size_t)T; t += stride) {
        flags[t] = ((t >= 4) && ((t & 1) == 0)) ? 1.0f : 0.0f;
    }
}

// ---------------------------------------------------------------------------
// WMMA GEMM: M x 160 x 64, fp32.
//   A[m][k] = x[64*m + k]        (overlapping windows, 160 contiguous floats)
//   B[k][n] = W[k*64 + n]        (staged in LDS, 40 KB)
//   out row t = 4 + 2*m  (+ zero row t+1 written from the same lane)
// Block: 256 threads = 8 waves; wave -> (mpair = w>>2, ntile = w&3).
// Each block covers ROWS_PER_BLOCK rows via 4 outer iterations of 32 rows.
// ---------------------------------------------------------------------------
__global__ __launch_bounds__(256) void conv_stream_gemm(const float* __restrict__ x,
                                                        const float* __restrict__ W,
                                                        float* __restrict__ out,
                                                        int M, int T) {
    __shared__ float sW[KTOT * COUT];   // 40 KB (<< 320 KB per WGP)

    const int tid = threadIdx.x;

    // Cooperative stage of W into LDS (contiguous float4 copy).
    {
        const float4* Wv = (const float4*)W;
        float4* sWv = (float4*)sW;
        #pragma unroll
        for (int i = tid; i < (KTOT * COUT) / 4; i += 256) sWv[i] = Wv[i];
    }
    __syncthreads();

    const int lane  = tid & 31;
    const int wave  = tid >> 5;
    const int g     = lane >> 4;        // lane group: 0 = lanes 0-15, 1 = 16-31
    const int ln    = lane & 15;
    const int ntile = wave & 3;         // 4 tiles across N=64
    const int mpair = wave >> 2;        // 2 M-tiles in flight per iteration
    const int n0    = ntile * 16;

    #pragma unroll
    for (int it = 0; it < 4; ++it) {
        const int m0 = blockIdx.x * ROWS_PER_BLOCK + it * 32 + mpair * 16;

        // Row this lane sources for the A fragment (clamped so tail blocks
        // never read past the input buffer; stores are guarded below).
        const int mrow  = m0 + ln;
        const int mload = (mrow < M) ? mrow : (M - 1);
        const float* __restrict__ arow = x + (size_t)(2 * CIN) * mload;

        v8f c = {};  // fp32 accumulator, 8 VGPRs

        #pragma unroll 8
        for (int k0 = 0; k0 < KTOT; k0 += 4) {
            const int ka = k0 + 2 * g;  // 32-bit A/B layout: VGPR j <-> K = 2g+j

            // A fragment: 16x4 fp32 (2 VGPRs), 8-byte-aligned b64 load
            v2f a;
            a.x = arow[ka];
            a.y = arow[ka + 1];

            // B fragment: 4x16 fp32 (2 VGPRs) from LDS
            v2f b;
            b.x = sW[ka * COUT + n0 + ln];
            b.y = sW[(ka + 1) * COUT + n0 + ln];

            // D = A*B + C  (v_wmma_f32_16x16x4_f32)
            c = __builtin_amdgcn_wmma_f32_16x16x4_f32(
                    /*neg_a=*/false, a, /*neg_b=*/false, b,
                    /*c_mod=*/(short)0, c, /*reuse_a=*/false, /*reuse_b=*/false);
        }

        // C/D layout: VGPR r, lane group g -> row m = m0 + r + 8g, col = ln.
        // Result row t = 4+2m; companion zero row t+1 (odd -> flag false).
        const int t0 = 4 + 2 * m0;      // first output row of this wave tile
        if ((m0 + 16 <= M) && (t0 + 31 < T)) {
            // Fast path: whole tile in range -> one base, immediate offsets.
            float* __restrict__ base =
                out + (size_t)(t0 + 16 * g) * COUT + n0 + ln;
            #pragma unroll
            for (int r = 0; r < 8; ++r) {
                base[(size_t)(2 * r) * COUT]     = c[r];   // even row (result)
                base[(size_t)(2 * r + 1) * COUT] = 0.0f;   // odd row (zero)
            }
        } else {
            #pragma unroll
            for (int r = 0; r < 8; ++r) {
                const int m = m0 + r + 8 * g;
                if (m < M) {
                    const size_t t = (size_t)(4 + 2 * m);
                    out[t * COUT + n0 + ln] = c[r];
                    if (t + 1 < (size_t)T) out[(t + 1) * COUT + n0 + ln] = 0.0f;
                }
            }
        }
    }
}

// ---------------------------------------------------------------------------
extern "C" void kernel_launch(void* const* d_in, const int* in_sizes, int n_in,
                              void* d_out, int out_size, void* d_ws, size_t ws_size,
                              hipStream_t stream) {
    const float* x = (const float*)d_in[0];   // [T, 32] fp32
    const float* W = (const float*)d_in[1];   // [5, 32, 64] fp32 (== [160,64] row-major)
    float* out = (float*)d_out;               // [T,64] ys then [T] flags

    const int T = in_sizes[0] / CIN;
    float* flags = out + (size_t)T * COUT;

    // Number of flagged rows: even t in [4, T-1]
    const int M = (T >= 5) ? ((T - 5) / 2 + 1) : 0;

    // 1) zero rows 0..3 (or everything if no GEMM rows) + flags for all rows
    fill_head_and_flags<<<512, 256, 0, stream>>>(out, flags, T, (M == 0) ? 1 : 0);

    // 2) WMMA GEMM: writes all rows t >= 4 (results on even t, zeros on odd t)
    if (M > 0) {
        const int grid = (M + ROWS_PER_BLOCK - 1) / ROWS_PER_BLOCK;
        conv_stream_gemm<<<grid, 256, 0, stream>>>(x, W, out, M, T);
    }
}